// Model_9938554323202
// MI455X (gfx1250) — compile-verified
//
#include <hip/hip_runtime.h>
#include <hip/hip_bf16.h>
#include <math.h>

typedef __attribute__((ext_vector_type(2))) float v2f;
typedef __attribute__((ext_vector_type(8))) float v8f;

// ---------------------------------------------------------------------------
// Problem constants
// ---------------------------------------------------------------------------
#define BS 64
#define TT 50
#define DD 9
#define NBASIS 55          // 1 + 9 + 45
#define NSYS   (BS * DD)   // 576
#define NK     151         // 3T+1
#define RDIM   300         // (3T+1) + (3T-1)
#define NCAUG  351         // 300 + 51 augmented columns
#define K1P    452         // 450 padded to multiple of 4
#define N3P    512         // 450 padded to multiple of 64
#define VSTRIDE 512        // var row stride in padded buffer

// output layout (floats)
#define OFF_X0    0        // 64*50*9  = 28800
#define OFF_STEPS 28800    // 64*9*49  = 28224
#define OFF_EPS   57024    // 64*9     = 576
#define OFF_VAR   57600    // 64*50*9  = 28800
#define OFF_XI    86400    // 55*9     = 495

// ---------------------------------------------------------------------------
// Zero-padded 2D copy:  dst(rowsD x colsD) <- src(rowsS x colsS), rest = 0
// ---------------------------------------------------------------------------
__global__ void pad2d(float* __restrict__ dst, const float* __restrict__ src,
                      int rowsD, int colsD, int rowsS, int colsS) {
    int i = blockIdx.x * blockDim.x + threadIdx.x;
    if (i >= rowsD * colsD) return;
    int r = i / colsD, c = i % colsD;
    dst[i] = (r < rowsS && c < colsS) ? src[r * colsS + c] : 0.0f;
}

// ---------------------------------------------------------------------------
// Batch-1 GEMV for the p-MLP (pure bandwidth, coalesced weight reads).
// ---------------------------------------------------------------------------
__global__ void gemv_relu(const float* __restrict__ x, const float* __restrict__ W,
                          const float* __restrict__ b, float* __restrict__ y,
                          int fi, int fo, int relu) {
    int o = blockIdx.x * blockDim.x + threadIdx.x;
    if (o >= fo) return;
    float acc = b[o];
    for (int i = 0; i < fi; ++i) acc += x[i] * W[i * fo + o];
    if (relu) acc = fmaxf(acc, 0.0f);
    y[o] = acc;
}

// ---------------------------------------------------------------------------
// fp32 WMMA GEMM, compile-time shapes, no edge handling (inputs pre-padded):
//   C(64 x N) = relu?(A(64 x K) @ W(K x N) + bias)
// block = 128 threads = 4 waves; wave w owns C rows [16w,16w+16) and computes
// a 16x64 strip (4 N-subtiles): one A fragment (b64 load) feeds 4 WMMAs.
// K, N constant => immediate address offsets, no masks, no u64 adds per load.
// ---------------------------------------------------------------------------
template <int K, int N>
__global__ void mlp_gemm_wmma(const float* __restrict__ A, const float* __restrict__ W,
                              const float* __restrict__ bias, float* __restrict__ C,
                              int relu) {
    const int lane  = threadIdx.x & 31;
    const int wave  = threadIdx.x >> 5;
    const int m0    = wave * 16;
    const int nbase = blockIdx.x * 64;
    const int mrow  = lane & 15;   // A fragment: M index
    const int khalf = lane >> 4;   // selects K pair {0,1} vs {2,3}
    const int ncol  = lane & 15;   // B/C fragment: N index

    const int nb0 = nbase + ncol;
    const int nb1 = nb0 + 16;
    const int nb2 = nb0 + 32;
    const int nb3 = nb0 + 48;

    const float* __restrict__ Arow = A + (m0 + mrow) * K;
    const float* __restrict__ Wl   = W + 2 * khalf * N;   // lane's K-pair base

    v8f c0 = {}, c1 = {}, c2 = {}, c3 = {};

#pragma unroll 2
    for (int k = 0; k < K; k += 4) {
        v2f a = *(const v2f*)(Arow + k + 2 * khalf);      // one b64 load
        const float* __restrict__ W0 = Wl + k * N;        // row ka
        const float* __restrict__ W1 = W0 + N;            // row ka+1
        v2f b0, b1, b2, b3;
        b0.x = W0[nb0];  b0.y = W1[nb0];
        b1.x = W0[nb1];  b1.y = W1[nb1];
        b2.x = W0[nb2];  b2.y = W1[nb2];
        b3.x = W0[nb3];  b3.y = W1[nb3];
        c0 = __builtin_amdgcn_wmma_f32_16x16x4_f32(false, a, false, b0, (short)0, c0, false, false);
        c1 = __builtin_amdgcn_wmma_f32_16x16x4_f32(false, a, false, b1, (short)0, c1, false, false);
        c2 = __builtin_amdgcn_wmma_f32_16x16x4_f32(false, a, false, b2, (short)0, c2, false, false);
        c3 = __builtin_amdgcn_wmma_f32_16x16x4_f32(false, a, false, b3, (short)0, c3, false, false);
    }

    // store: C layout lane l, vgpr v -> M = v + 8*(l>>4), N = l&15
    v8f acc[4] = {c0, c1, c2, c3};
#pragma unroll
    for (int s = 0; s < 4; ++s) {
        int nb = nb0 + 16 * s;
        float bv = bias[nb];
#pragma unroll
        for (int v = 0; v < 8; ++v) {
            int m = m0 + v + 8 * khalf;
            float val = acc[s][v] + bv;
            if (relu) val = fmaxf(val, 0.0f);
            C[m * N + nb] = val;
        }
    }
}

// ---------------------------------------------------------------------------
// rhs[b,d,t] = sum_k library(var[b,t,:])[k] * xi[k,d]   (fp32, stored as fp64)
// var is read from the padded buffer (row stride VSTRIDE).
// ---------------------------------------------------------------------------
__global__ void rhs_kernel(const float* __restrict__ varp, const float* __restrict__ xi,
                           double* __restrict__ rhs) {
    int inst = blockIdx.x;           // b*9 + d
    int b = inst / DD, d = inst % DD;
    int t = threadIdx.x;
    if (t >= TT) return;
    const float* v = varp + b * VSTRIDE + t * DD;
    float vv[DD];
#pragma unroll
    for (int i = 0; i < DD; ++i) vv[i] = v[i];
    float acc = xi[d];                                   // constant basis, k=0
#pragma unroll
    for (int i = 0; i < DD; ++i) acc += vv[i] * xi[(1 + i) * DD + d];
    int k = 1 + DD;
#pragma unroll
    for (int i = 0; i < DD; ++i)
        for (int j = i; j < DD; ++j, ++k)
            acc += (vv[i] * vv[j]) * xi[k * DD + d];     // triu quad terms
    rhs[inst * TT + t] = (double)acc;
}

// ---------------------------------------------------------------------------
// steps / var / xi passthrough outputs (var read through padded stride)
// ---------------------------------------------------------------------------
__global__ void write_misc(const float* __restrict__ varp, const float* __restrict__ xi,
                           float* __restrict__ out) {
    int i = blockIdx.x * blockDim.x + threadIdx.x;
    double l  = log(0.01 / 0.99);
    float  st = (float)(1.0 / (1.0 + exp(-l)));          // sigmoid(logit(0.01)) = 0.01
    if (i < BS * TT * DD) {
        int b = i / (TT * DD), c = i % (TT * DD);
        out[OFF_VAR + i] = varp[b * VSTRIDE + c];
    }
    if (i < BS * DD * (TT - 1)) out[OFF_STEPS + i] = st;
    if (i < NBASIS * DD) out[OFF_XI + i] = xi[i];
}

// ---------------------------------------------------------------------------
// Build augmented [K | E] (300 x 351) fp64.  K = [[Q, A^T],[A, 0]] with
// c = [0,1,0] and h = 0.01 identical for every instance -> built ONCE.
// E = the 51 unit columns e_{151..201} hit by b = [rhs(50), iv, 0...].
// ---------------------------------------------------------------------------
__device__ inline void setA(double* M, int r, int c, double v) {
    M[(NK + r) * NCAUG + c] = v;     // A block
    M[c * NCAUG + (NK + r)] = v;     // A^T block
}

__global__ void build_K(double* __restrict__ M) {
    for (int i = threadIdx.x; i < RDIM * NCAUG; i += blockDim.x) M[i] = 0.0;
    __syncthreads();
    if (threadIdx.x != 0) return;
    const double h = 1.0 / (1.0 + exp(-log(0.01 / 0.99)));   // = 0.01
    for (int i = 0; i < 3 * TT; ++i) M[i * NCAUG + i] = 0.05;
    M[(3 * TT) * NCAUG + (3 * TT)] = 1.0;
    for (int i = 0; i < TT; ++i) setA(M, i, TT + i, 1.0);    // c1 = 1 rows
    setA(M, TT, 0, 1.0);                                     // initial value
    for (int j = 0; j < TT - 1; ++j) {
        int r0 = TT + 1 + j;
        setA(M, r0, j + 1, 1.0);
        setA(M, r0, j, -1.0);
        setA(M, r0, TT + j, -h);
        setA(M, r0, 2 * TT + j, -0.5 * h * h);
        setA(M, r0, 3 * TT, -0.5 * h * h);
        int r1 = 2 * TT + j;
        setA(M, r1, TT + j + 1, 1.0);
        setA(M, r1, TT + j, -1.0);
        setA(M, r1, 2 * TT + j, -h);
        setA(M, r1, 3 * TT, -h);
    }
    for (int j = 0; j <= TT; ++j) M[(NK + j) * NCAUG + RDIM + j] = 1.0;   // E
}

// ---------------------------------------------------------------------------
// One-shot Gauss-Jordan on [K | E] in a single workgroup (no pivoting needed:
// Q pivots are 0.05/1.0 and the Schur complement -A Q^-1 A^T is definite).
// Leaves Y = K^-1 E in columns 300..350.
// ---------------------------------------------------------------------------
__global__ void gauss_jordan(double* __restrict__ M) {
    __shared__ double colp[RDIM];
    __shared__ double prow[NCAUG];
    __shared__ double pinv;
    for (int p = 0; p < RDIM; ++p) {
        if (threadIdx.x == 0) pinv = 1.0 / M[p * NCAUG + p];
        __syncthreads();
        for (int c = threadIdx.x; c < NCAUG; c += blockDim.x) {
            double v = M[p * NCAUG + c] * pinv;
            M[p * NCAUG + c] = v;
            prow[c] = v;
        }
        for (int r = threadIdx.x; r < RDIM; r += blockDim.x) colp[r] = M[r * NCAUG + p];
        __syncthreads();
        for (int idx = threadIdx.x; idx < RDIM * NCAUG; idx += blockDim.x) {
            int r = idx / NCAUG;
            if (r != p) M[idx] -= colp[r] * prow[idx - r * NCAUG];
        }
        __syncthreads();
    }
}

// ---------------------------------------------------------------------------
// Per-instance solve: u[row] = sum_j Y[row][j]*b[j], only rows 0..49 (x0)
// and row 150 (eps) are needed.  576 blocks x 64 threads.
// ---------------------------------------------------------------------------
__global__ void solve_out(const double* __restrict__ M, const double* __restrict__ rhs,
                          const float* __restrict__ varp, float* __restrict__ out) {
    int inst = blockIdx.x;               // b*9 + d
    int b = inst / DD, d = inst % DD;
    __shared__ double rb[TT + 1];
    int t = threadIdx.x;
    if (t < TT) rb[t] = rhs[inst * TT + t];
    if (t == TT) rb[TT] = (double)varp[b * VSTRIDE + d];  // iv = var[b,0,d]
    __syncthreads();
    if (t <= TT) {
        int row = (t < TT) ? t : 3 * TT;                  // x0 rows, then eps row
        const double* __restrict__ Y = M + row * NCAUG + RDIM;
        double acc = 0.0;
#pragma unroll 8
        for (int j = 0; j <= TT; ++j) acc += Y[j] * rb[j];
        if (t < TT) out[OFF_X0 + b * TT * DD + t * DD + d] = (float)acc;
        else        out[OFF_EPS + b * DD + d]              = (float)acc;
    }
}

// ---------------------------------------------------------------------------
// Launch
// ---------------------------------------------------------------------------
extern "C" void kernel_launch(void* const* d_in, const int* in_sizes, int n_in,
                              void* d_out, int out_size, void* d_ws, size_t ws_size,
                              hipStream_t stream) {
    (void)in_sizes; (void)n_in; (void)out_size; (void)ws_size;
    // inputs: 0 index, 1 net_iv, 2 p_in, 3 pW1, 4 pb1, 5 pW2, 6 pb2, 7 pW3,
    //         8 pb3, 9 nW1, 10 nb1, 11 nW2, 12 nb2, 13 nW3, 14 nb3
    const float* net_iv = (const float*)d_in[1];
    const float* p_in   = (const float*)d_in[2];
    const float* pW1 = (const float*)d_in[3];  const float* pb1 = (const float*)d_in[4];
    const float* pW2 = (const float*)d_in[5];  const float* pb2 = (const float*)d_in[6];
    const float* pW3 = (const float*)d_in[7];  const float* pb3 = (const float*)d_in[8];
    const float* nW1 = (const float*)d_in[9];  const float* nb1 = (const float*)d_in[10];
    const float* nW2 = (const float*)d_in[11]; const float* nb2 = (const float*)d_in[12];
    const float* nW3 = (const float*)d_in[13]; const float* nb3 = (const float*)d_in[14];
    float* out = (float*)d_out;

    char* ws = (char*)d_ws;
    float*  ph1  = (float*)(ws + 0);              // 1024 f32
    float*  ph2  = (float*)(ws + 4096);           // 1024 f32
    float*  xi   = (float*)(ws + 8192);           // 495 f32
    float*  nh1  = (float*)(ws + 10240);          // 64x1024 f32
    float*  nh2  = (float*)(ws + 272384);         // 64x1024 f32
    float*  varp = (float*)(ws + 534528);         // 64x512 f32 (padded var)
    double* rhs  = (double*)(ws + 665600);        // 576x50 f64
    double* Maug = (double*)(ws + 896000);        // 300x351 f64
    float*  A1p  = (float*)(ws + 1738400);        // 64x452 f32
    float*  W1p  = (float*)(ws + 1854112);        // 452x1024 f32
    float*  W3p  = (float*)(ws + 3705504);        // 1024x512 f32
    float*  b3p  = (float*)(ws + 5802656);        // 512 f32

    // zero-pad A1 / W1 / W3 / b3 so WMMA GEMMs need no edge handling
    pad2d<<<(64 * K1P + 255) / 256, 256, 0, stream>>>(A1p, net_iv, 64, K1P, 64, 450);
    pad2d<<<(K1P * 1024 + 255) / 256, 256, 0, stream>>>(W1p, nW1, K1P, 1024, 450, 1024);
    pad2d<<<(1024 * N3P + 255) / 256, 256, 0, stream>>>(W3p, nW3, 1024, N3P, 1024, 450);
    pad2d<<<(N3P + 255) / 256, 256, 0, stream>>>(b3p, nb3, 1, N3P, 1, 450);

    // p-MLP (batch 1) -> xi
    gemv_relu<<<4, 256, 0, stream>>>(p_in, pW1, pb1, ph1, 64, 1024, 1);
    gemv_relu<<<4, 256, 0, stream>>>(ph1, pW2, pb2, ph2, 1024, 1024, 1);
    gemv_relu<<<2, 256, 0, stream>>>(ph2, pW3, pb3, xi, 1024, NBASIS * DD, 0);

    // n-MLP (M = 64) via fp32 WMMA, all shapes compile-time and padded
    mlp_gemm_wmma<K1P, 1024><<<16, 128, 0, stream>>>(A1p, W1p, nb1, nh1, 1);
    mlp_gemm_wmma<1024, 1024><<<16, 128, 0, stream>>>(nh1, nW2, nb2, nh2, 1);
    mlp_gemm_wmma<1024, N3P><<<N3P / 64, 128, 0, stream>>>(nh2, W3p, b3p, varp, 0);

    // library + einsum -> rhs (fp64), passthrough outputs
    rhs_kernel<<<NSYS, 64, 0, stream>>>(varp, xi, rhs);
    write_misc<<<(BS * TT * DD + 255) / 256, 256, 0, stream>>>(varp, xi, out);

    // single shared KKT factorization, then 576 tiny matvecs
    build_K<<<1, 256, 0, stream>>>(Maug);
    gauss_jordan<<<1, 1024, 0, stream>>>(Maug);
    solve_out<<<NSYS, 64, 0, stream>>>(Maug, rhs, varp, out);
}